// ModelNew_25056839204948
// MI455X (gfx1250) — compile-verified
//
#include <hip/hip_runtime.h>

// Masked cumulative sum along rows, B=256 x N=131072, fp32.
// Strategy: prefix-sum == lower-triangular matmul. Each wave scans a
// 256-element chunk as a 16(pos) x 16(seg) tile with 4 accumulated
// V_WMMA_F32_16X16X4_F32 ops (A = constant triangular block, full fp32).
// Cross-segment / cross-chunk / cross-wave carries via shuffles + LDS.

typedef __attribute__((ext_vector_type(2))) float v2f;
typedef __attribute__((ext_vector_type(8))) float v8f;

constexpr int BROWS = 256;
constexpr int NCOLS = 131072;
constexpr int WAVES = 8;        // 256-thread block = 8 wave32
constexpr int CPW   = 4;        // chunks (of 256 elems) per wave per iteration
constexpr int CHUNK = 256;
constexpr int EPW   = CPW * CHUNK;      // 1024 elems per wave per iter
constexpr int EPB   = WAVES * EPW;      // 8192 elems per block per iter
constexpr int ITERS = NCOLS / EPB;      // 16

__global__ __launch_bounds__(256) void masked_cumsum_wmma(
    const float* __restrict__ x,
    const unsigned char* __restrict__ mask,
    float* __restrict__ out)
{
    const int row  = blockIdx.x;
    const int tid  = threadIdx.x;
    const int wave = tid >> 5;
    const int lane = tid & 31;
    const int l15  = lane & 15;
    const int hi   = lane >> 4;          // half-wave select

    const float*         xr = x    + (size_t)row * NCOLS;
    const unsigned char* mr = mask + (size_t)row * NCOLS;
    float*               orow = out + (size_t)row * NCOLS;

    __shared__ float ldsTot[2][WAVES];   // ping-pong: 1 barrier per iteration

    // A-matrix blocks: L[m][k] = (k <= m), split into four 16x4 K-blocks.
    // A layout (32-bit 16x4): lane m (and m+16) hold row m; VGPR0/1 = K pair
    // (K=0,1 for lanes 0-15; K=2,3 for lanes 16-31).
    v2f a[4];
#pragma unroll
    for (int j = 0; j < 4; ++j) {
        const int k0 = 4 * j + 2 * hi;
        a[j].x = (k0     <= l15) ? 1.0f : 0.0f;
        a[j].y = (k0 + 1 <= l15) ? 1.0f : 0.0f;
    }

    float row_carry = 0.0f;              // uniform across block

    for (int it = 0; it < ITERS; ++it) {
        const int wave_base = it * EPB + wave * EPW;

        v8f   acc[CPW];
        float local = 0.0f;              // carry within this wave's 4 chunks

#pragma unroll
        for (int c = 0; c < CPW; ++c) {
            const int cbase = wave_base + c * CHUNK;
            v8f d = {};
            // B layout (32-bit 4x16): lane n (n+16) hold column n; VGPR0/1 =
            // K pair (K=0,1 lanes 0-15; K=2,3 lanes 16-31).
            // B_j[k][n] = masked x[cbase + n*16 + 4j + k].
#pragma unroll
            for (int j = 0; j < 4; ++j) {
                const int off = cbase + l15 * 16 + 4 * j + 2 * hi;
                v2f xv = *(const v2f*)(xr + off);                  // 8B aligned
                const unsigned short ms =
                    *(const unsigned short*)(mr + off);            // 2B aligned
                v2f b;
                b.x = (ms & 0x00FFu) ? xv.x : 0.0f;
                b.y = (ms & 0xFF00u) ? xv.y : 0.0f;
                // D[m][n] += sum_k (4j+k <= m) * x[n*16+4j+k]
                d = __builtin_amdgcn_wmma_f32_16x16x4_f32(
                        false, a[j], false, b, (short)0, d, false, false);
            }
            // d now holds intra-segment inclusive prefixes:
            // lane n (cols) rows hi*8..hi*8+7 in d[0..7].
            // Cross-segment carry inside the 256-element chunk:
            float tot  = __shfl(d[7], 16 + l15, 32);   // segment l15 total
            float incl = tot;
#pragma unroll
            for (int dl = 1; dl < 16; dl <<= 1) {
                float u = __shfl_up(incl, dl, 16);     // both halves identical
                if (l15 >= dl) incl += u;
            }
            const float ex          = incl - tot;      // exclusive seg prefix
            const float chunk_total = __shfl(incl, 15, 16);
            const float s = ex + local;
#pragma unroll
            for (int i = 0; i < 8; ++i) d[i] += s;
            acc[c] = d;
            local += chunk_total;                      // uniform per lane
        }

        // Exchange per-wave iteration totals (single barrier, ping-pong LDS).
        if (lane == 0) ldsTot[it & 1][wave] = local;
        __syncthreads();
        float wave_off = 0.0f, iter_total = 0.0f;
#pragma unroll
        for (int w = 0; w < WAVES; ++w) {
            const float t = ldsTot[it & 1][w];
            if (w < wave) wave_off += t;
            iter_total += t;
        }

        const float base = row_carry + wave_off;
#pragma unroll
        for (int c = 0; c < CPW; ++c) {
            v8f d = acc[c];
#pragma unroll
            for (int i = 0; i < 8; ++i) d[i] += base;
            const int cbase = wave_base + c * CHUNK;
            // lane owns elements cbase + l15*16 + hi*8 .. +7 (32B aligned)
            *(v8f*)(orow + cbase + l15 * 16 + hi * 8) = d;
        }
        row_carry += iter_total;
    }
}

extern "C" void kernel_launch(void* const* d_in, const int* in_sizes, int n_in,
                              void* d_out, int out_size, void* d_ws, size_t ws_size,
                              hipStream_t stream) {
    const float*         x    = (const float*)d_in[0];
    const unsigned char* mask = (const unsigned char*)d_in[1];  // jnp.bool_: 1B/elem
    float*               out  = (float*)d_out;
    (void)in_sizes; (void)n_in; (void)out_size; (void)d_ws; (void)ws_size;

    dim3 grid(BROWS), block(WAVES * 32);
    hipLaunchKernelGGL(masked_cumsum_wmma, grid, block, 0, stream, x, mask, out);
}